// QuantumAttentionLayer_14594298872239
// MI455X (gfx1250) — compile-verified
//
#include <hip/hip_runtime.h>
#include <math.h>

typedef __attribute__((ext_vector_type(2))) float v2f;
typedef __attribute__((ext_vector_type(8))) float v8f;

#define PI_F 3.14159265358979323846f

static __device__ __forceinline__ v8f wmma4(v2f a, v2f b, v8f c) {
  // V_WMMA_F32_16X16X4_F32: full fp32 matrix op, D = A(16x4) x B(4x16) + C
  return __builtin_amdgcn_wmma_f32_16x16x4_f32(false, a, false, b, (short)0, c,
                                               false, false);
}

// ---------------------------------------------------------------------------
// Generic strided WMMA GEMM: C[z] = A[z] @ B[z] (+ bias), fp32 throughout.
// Block = 256 threads = 8 wave32 waves; block tile 64x64, wave tile 16x32.
// Batch offsets: off(z) = (z/zdiv)*zso + (z%zdiv)*zsi  (handles (b,h) splits).
// Requires M,N multiples of 64 and K multiple of 32 (true for every use here).
// ---------------------------------------------------------------------------
__global__ __launch_bounds__(256) void gemm_wmma_f32(
    const float* __restrict__ A, const float* __restrict__ B,
    const float* __restrict__ bias, float* __restrict__ C, int K,
    long rsA, long csA, long rsB, long csB, long rsC, long csC,
    int zdiv, long zsAo, long zsAi, long zsBo, long zsBi, long zsCo, long zsCi)
{
  int z = blockIdx.z;
  const float* Ab = A + (long)(z / zdiv) * zsAo + (long)(z % zdiv) * zsAi;
  const float* Bb = B + (long)(z / zdiv) * zsBo + (long)(z % zdiv) * zsBi;
  float*       Cb = C + (long)(z / zdiv) * zsCo + (long)(z % zdiv) * zsCi;
  long m0 = (long)blockIdx.y * 64;
  long n0 = (long)blockIdx.x * 64;

  __shared__ float As[64][33];   // 64xBK(=32), padded: odd stride kills conflicts
  __shared__ float Bs[32][65];   // BKx64, padded

  int tid  = threadIdx.x;
  int lane = tid & 31;
  int w    = tid >> 5;
  int wr   = w >> 1;             // 0..3 -> M sub-tile
  int wc   = w & 1;              // 0..1 -> N sub-tile
  int l16  = lane & 15;
  int khalf = (lane >> 4) << 1;  // f32 frag: K = 2*(lane/16) + vgpr

  v8f c0 = {0.f, 0.f, 0.f, 0.f, 0.f, 0.f, 0.f, 0.f};
  v8f c1 = c0;

  for (int k0 = 0; k0 < K; k0 += 32) {
    {
      int r = tid >> 3;
      int c = (tid & 7) << 2;
#pragma unroll
      for (int half = 0; half < 2; ++half) {
        int rr = r + half * 32;
        long base = (m0 + rr) * rsA + (long)(k0 + c) * csA;
#pragma unroll
        for (int j = 0; j < 4; ++j) As[rr][c + j] = Ab[base + j * csA];
      }
    }
    {
      int r = tid >> 4;
      int c = (tid & 15) << 2;
#pragma unroll
      for (int half = 0; half < 2; ++half) {
        int rr = r + half * 16;
        long base = (long)(k0 + rr) * rsB + (n0 + c) * csB;
#pragma unroll
        for (int j = 0; j < 4; ++j) Bs[rr][c + j] = Bb[base + j * csB];
      }
    }
    __syncthreads();

    int mrow = wr * 16 + l16;
    int nb0  = wc * 32 + l16;
#pragma unroll
    for (int kk = 0; kk < 32; kk += 4) {
      v2f a, b0, b1;
      a.x  = As[mrow][kk + khalf];
      a.y  = As[mrow][kk + khalf + 1];
      b0.x = Bs[kk + khalf][nb0];
      b0.y = Bs[kk + khalf + 1][nb0];
      b1.x = Bs[kk + khalf][nb0 + 16];
      b1.y = Bs[kk + khalf + 1][nb0 + 16];
      c0 = wmma4(a, b0, c0);
      c1 = wmma4(a, b1, c1);
    }
    __syncthreads();
  }

  // C/D layout: vgpr i, lane l -> M = 8*(l/16)+i, N = l&15
  long mbase = m0 + wr * 16 + (long)(lane >> 4) * 8;
  long nc0 = n0 + wc * 32 + l16;
  long nc1 = nc0 + 16;
  float bb0 = bias ? bias[nc0] : 0.f;
  float bb1 = bias ? bias[nc1] : 0.f;
#pragma unroll
  for (int i = 0; i < 8; ++i) {
    Cb[(mbase + i) * rsC + nc0 * csC] = c0[i] + bb0;
    Cb[(mbase + i) * rsC + nc1 * csC] = c1[i] + bb1;
  }
}

// ---------------------------------------------------------------------------
// Quantum scalar path: only columns 0..7 of (feat @ We + be) matter; the
// superposition collapses to a single signed amplitude per stream.
// One block per (b,s) row. Writes scale[b*S+s]; atomically accumulates the
// coherence / entropy sums.
// ---------------------------------------------------------------------------
__global__ __launch_bounds__(256) void quantum_kernel(
    const float* __restrict__ audio, const float* __restrict__ visual,
    const float* __restrict__ We, const float* __restrict__ be,
    float* __restrict__ scale, float* __restrict__ accums)
{
  int row = blockIdx.x;                       // 0..511  (= b*256 + s)
  const float* a = audio  + (long)row * 512;
  const float* v = visual + (long)row * 512;
  int t = threadIdx.x;

  float pa[8], pv[8];
#pragma unroll
  for (int q = 0; q < 8; ++q) { pa[q] = 0.f; pv[q] = 0.f; }
  for (int k = t; k < 512; k += 256) {
    float av_ = a[k], vv_ = v[k];
    const float* wrow = We + (long)k * 512;
#pragma unroll
    for (int q = 0; q < 8; ++q) {
      float wq = wrow[q];
      pa[q] += av_ * wq;
      pv[q] += vv_ * wq;
    }
  }

  __shared__ float red[256];
  __shared__ float eres[16];
  for (int j = 0; j < 16; ++j) {
    red[t] = (j < 8) ? pa[j] : pv[j - 8];
    __syncthreads();
    for (int s = 128; s > 0; s >>= 1) {
      if (t < s) red[t] += red[t + s];
      __syncthreads();
    }
    if (t == 0) eres[j] = red[0] + be[j & 7];
    __syncthreads();
  }

  if (t == 0) {
    float ca = 1.f, cv = 1.f;
#pragma unroll
    for (int q = 0; q < 8; ++q) {
      ca *= cosf(0.5f * PI_F * eres[q]);
      cv *= cosf(0.5f * PI_F * eres[8 + q]);
    }
    float sa = ca / (fabsf(ca) + 1e-10f);
    float sv = cv / (fabsf(cv) + 1e-10f);
    float e  = 0.7071f * sa * sv;             // both nonzero ent entries
    float e2 = e * e;
    float qsum = 2.f * e2;
    float interference = (e < 0.f) ? cosf(2.f * PI_F / 65536.f) : 1.f;
    scale[row] = qsum * interference;
    atomicAdd(&accums[0], 2.f * fabsf(e));    // sum |ent|
    float p = e2 / (qsum + 1e-10f);
    atomicAdd(&accums[1], -2.f * p * logf(p + 1e-10f));
  }
}

// Row softmax with the per-(b,q) quantum scale folded in (scale >= 0, so
// softmax(alpha*x) is the reference logits exactly). Head 7 rows are also
// streamed to the attn[:, -1] output.
__global__ __launch_bounds__(256) void softmax_kernel(
    float* __restrict__ attn, const float* __restrict__ scale,
    float* __restrict__ out_attn)
{
  int idx = blockIdx.x;                       // (b*8+h)*256 + q
  int q = idx & 255;
  int h = (idx >> 8) & 7;
  int b = idx >> 11;
  float* rowp = attn + (long)idx * 256;
  int t = threadIdx.x;

  float alpha = scale[b * 256 + q] * 0.125f;  // scale / sqrt(64)
  float vl = rowp[t] * alpha;

  __shared__ float red[256];
  red[t] = vl;
  __syncthreads();
  for (int s = 128; s > 0; s >>= 1) {
    if (t < s) red[t] = fmaxf(red[t], red[t + s]);
    __syncthreads();
  }
  float mx = red[0];
  __syncthreads();
  float ex = expf(vl - mx);
  red[t] = ex;
  __syncthreads();
  for (int s = 128; s > 0; s >>= 1) {
    if (t < s) red[t] += red[t + s];
    __syncthreads();
  }
  float p = ex / red[0];
  rowp[t] = p;
  if (h == 7) out_attn[((long)b * 256 + q) * 256 + t] = p;
}

// Residual + LayerNorm; rows 0..511 audio, 512..1023 visual.
__global__ __launch_bounds__(256) void layernorm_kernel(
    const float* __restrict__ audio, const float* __restrict__ visual,
    const float* __restrict__ proj_a, const float* __restrict__ proj_v,
    const float* __restrict__ gamma, const float* __restrict__ beta,
    float* __restrict__ out)
{
  int r = blockIdx.x;
  bool isA = r < 512;
  int row = isA ? r : r - 512;
  const float* f = (isA ? audio  : visual) + (long)row * 512;
  const float* p = (isA ? proj_a : proj_v) + (long)row * 512;
  float* o = out + (isA ? 0L : 262144L) + (long)row * 512;
  int t = threadIdx.x;

  float x0 = f[t] + p[t];
  float x1 = f[t + 256] + p[t + 256];

  __shared__ float red[256];
  red[t] = x0 + x1;
  __syncthreads();
  for (int s = 128; s > 0; s >>= 1) {
    if (t < s) red[t] += red[t + s];
    __syncthreads();
  }
  float mean = red[0] * (1.f / 512.f);
  __syncthreads();
  float d0 = x0 - mean, d1 = x1 - mean;
  red[t] = d0 * d0 + d1 * d1;
  __syncthreads();
  for (int s = 128; s > 0; s >>= 1) {
    if (t < s) red[t] += red[t + s];
    __syncthreads();
  }
  float inv = rsqrtf(red[0] * (1.f / 512.f) + 1e-5f);
  o[t]       = d0 * inv * gamma[t]       + beta[t];
  o[t + 256] = d1 * inv * gamma[t + 256] + beta[t + 256];
}

__global__ void init_accums_kernel(float* accums) {
  accums[0] = 0.f;
  accums[1] = 0.f;
}

__global__ void finalize_kernel(const float* __restrict__ accums,
                                float* __restrict__ out) {
  out[524288] = accums[0] / (512.f * 65536.f);  // quantum_coherence
  out[524289] = accums[1] / 512.f;              // entanglement_entropy
}

// ---------------------------------------------------------------------------
extern "C" void kernel_launch(void* const* d_in, const int* in_sizes, int n_in,
                              void* d_out, int out_size, void* d_ws,
                              size_t ws_size, hipStream_t stream) {
  const float* audio  = (const float*)d_in[0];
  const float* visual = (const float*)d_in[1];
  const float* Wq = (const float*)d_in[2];
  const float* bq = (const float*)d_in[3];
  const float* Wk = (const float*)d_in[4];
  const float* bk = (const float*)d_in[5];
  const float* Wv = (const float*)d_in[6];
  const float* bv = (const float*)d_in[7];
  const float* Wo = (const float*)d_in[8];
  const float* bo = (const float*)d_in[9];
  const float* We = (const float*)d_in[10];
  const float* be = (const float*)d_in[11];
  const float* gamma = (const float*)d_in[12];
  const float* beta  = (const float*)d_in[13];
  float* out = (float*)d_out;

  float* ws   = (float*)d_ws;
  float* aq   = ws;                 // 512x512
  float* av   = ws + 262144;        // 512x512
  float* vk   = ws + 524288;        // 512x512
  float* vv   = ws + 786432;        // 512x512
  float* attn = ws + 1048576;       // 16 x 256 x 256 (logits -> probs in place)
  float* aatt = ws + 2097152;       // [B,S,D]
  float* vatt = ws + 2359296;       // [B,S,D]
  float* scl  = ws + 2621440;       // 512
  float* accums = ws + 2621952;     // 2
  float* proj_a = aq;               // aq dead after logits GEMM
  float* proj_v = av;               // av dead after visual_att GEMM

  auto gemm = [&](const float* A, const float* B, const float* bias, float* C,
                  int M, int N, int K, long rsA, long csA, long rsB, long csB,
                  long rsC, long csC, int nz, int zdiv, long zsAo, long zsAi,
                  long zsBo, long zsBi, long zsCo, long zsCi) {
    dim3 grid(N / 64, M / 64, nz);
    gemm_wmma_f32<<<grid, 256, 0, stream>>>(A, B, bias, C, K, rsA, csA, rsB,
                                            csB, rsC, csC, zdiv, zsAo, zsAi,
                                            zsBo, zsBi, zsCo, zsCi);
  };

  init_accums_kernel<<<1, 1, 0, stream>>>(accums);
  quantum_kernel<<<512, 256, 0, stream>>>(audio, visual, We, be, scl, accums);

  // Q/K/V projections (fp32 WMMA, 512^3 each)
  gemm(audio,  Wq, bq, aq, 512, 512, 512, 512, 1, 512, 1, 512, 1,
       1, 1, 0, 0, 0, 0, 0, 0);
  gemm(audio,  Wv, bv, av, 512, 512, 512, 512, 1, 512, 1, 512, 1,
       1, 1, 0, 0, 0, 0, 0, 0);
  gemm(visual, Wk, bk, vk, 512, 512, 512, 512, 1, 512, 1, 512, 1,
       1, 1, 0, 0, 0, 0, 0, 0);
  gemm(visual, Wv, bv, vv, 512, 512, 512, 512, 1, 512, 1, 512, 1,
       1, 1, 0, 0, 0, 0, 0, 0);

  // logits[z=(b,h)] = aqh @ vkh^T  (B read with swapped strides)
  gemm(aq, vk, nullptr, attn, 256, 256, 64, 512, 1, 1, 512, 256, 1,
       16, 8, 131072, 64, 131072, 64, 524288, 65536);

  softmax_kernel<<<4096, 256, 0, stream>>>(attn, scl, out + 524290);

  // audio_att = attn @ vvh ; visual_att = attn^T @ avh (written in [B,S,D])
  gemm(attn, vv, nullptr, aatt, 256, 64, 256, 256, 1, 512, 1, 512, 1,
       16, 8, 524288, 65536, 131072, 64, 131072, 64);
  gemm(attn, av, nullptr, vatt, 256, 64, 256, 1, 256, 512, 1, 512, 1,
       16, 8, 524288, 65536, 131072, 64, 131072, 64);

  // output projections
  gemm(aatt, Wo, bo, proj_a, 512, 512, 512, 512, 1, 512, 1, 512, 1,
       1, 1, 0, 0, 0, 0, 0, 0);
  gemm(vatt, Wo, bo, proj_v, 512, 512, 512, 512, 1, 512, 1, 512, 1,
       1, 1, 0, 0, 0, 0, 0, 0);

  layernorm_kernel<<<1024, 256, 0, stream>>>(audio, visual, proj_a, proj_v,
                                             gamma, beta, out);
  finalize_kernel<<<1, 1, 0, stream>>>(accums, out);
}